// MixtureGaussianBatchLanguageModel_47588237640156
// MI455X (gfx1250) — compile-verified
//
#include <hip/hip_runtime.h>
#include <hip/hip_bf16.h>
#include <math.h>

#define DIM 10
#define NTOKN 2002
#define ICOMP 2
#define TCOMP 2
#define OCOMP 2
#define MAXC 4
#define BB 4
#define LL 16
#define NEGV (-1e30f)
#define LOG2PI_F 1.8378770664093453f

typedef __attribute__((ext_vector_type(2))) float v2f;
typedef __attribute__((ext_vector_type(8))) float v8f;

// ---------------------------------------------------------------------------
// In-place Gauss-Jordan inverse of the leading 10x10 block of a row-major
// matrix with row stride `st` in LDS. Returns log|det| (pivots > 0 for SPD).
// Padding outside the 10x10 block is never touched.
// ---------------------------------------------------------------------------
__device__ inline float gj_invert10s(float* M, const int st) {
  float logdet = 0.f;
  for (int p = 0; p < 10; ++p) {
    const float piv = M[p * st + p];
    logdet += __logf(fabsf(piv));
    const float inv = 1.f / piv;
    M[p * st + p] = 1.f;
    for (int j = 0; j < 10; ++j) M[p * st + j] *= inv;
    for (int i = 0; i < 10; ++i) {
      if (i == p) continue;
      const float f = M[i * st + p];
      M[i * st + p] = 0.f;
      for (int j = 0; j < 10; ++j) M[i * st + j] -= f * M[p * st + j];
    }
  }
  return logdet;
}
__device__ inline float gj_invert10(float* M) { return gj_invert10s(M, 10); }

// ---------------------------------------------------------------------------
// zeta = -0.5*(d*log(2pi) + logdet(cov + diag(dvar)) + diff^T (cov+diag)^-1 diff)
// via Cholesky; diff^T V^-1 diff == ||L^-1 diff||^2 (forward solve only).
// Fully unrolled so private arrays stay in VGPRs.
// ---------------------------------------------------------------------------
__device__ inline float gauss_zeta10(const float* cov, const float* dvar,
                                     const float* diff) {
  float L[10][10];
  float ld = 0.f;
#pragma unroll
  for (int p = 0; p < 10; ++p) {
    float s = cov[p * 10 + p] + dvar[p];
#pragma unroll
    for (int q = 0; q < p; ++q) s -= L[p][q] * L[p][q];
    const float lp = sqrtf(s);
    ld += __logf(lp);
    L[p][p] = lp;
    const float inv = 1.f / lp;
#pragma unroll
    for (int i = p + 1; i < 10; ++i) {
      float s2 = cov[i * 10 + p];
#pragma unroll
      for (int q = 0; q < p; ++q) s2 -= L[i][q] * L[p][q];
      L[i][p] = s2 * inv;
    }
  }
  float y[10];
  float quad = 0.f;
#pragma unroll
  for (int i = 0; i < 10; ++i) {
    float s = diff[i];
#pragma unroll
    for (int q = 0; q < i; ++q) s -= L[i][q] * y[q];
    y[i] = s / L[i][i];
    quad += y[i] * y[i];
  }
  return -0.5f * (10.f * LOG2PI_F + 2.f * ld + quad);
}

// ---------------------------------------------------------------------------
// Wave-cooperative 16x16 = 16x16 * 16x16 product via V_WMMA_F32_16X16X4_F32.
// All tiles are zero-padded 16x16 (stride 16) in LDS, so every load/store is
// unconditional (no EXEC masking around the LDS traffic). K is swept in 3
// chunks of 4 (K>=12 columns are zero padding) with accumulator chaining.
// A layout (16x4 f32): VGPR0 lanes0-15 K=0 / lanes16-31 K=2; VGPR1 K=1 / K=3.
// B layout (4x16 f32): mirrored. C/D: VGPR r = rows r / r+8 per lane half.
// Executed by a full wave, EXEC all ones (uniform control flow).
// ---------------------------------------------------------------------------
__device__ inline void wmma_mul16(float* dst, const float* X, const float* Y,
                                  int lane) {
  const int m = lane & 15;
  const int half = lane >> 4;
  v8f acc = {};
#pragma unroll
  for (int kc = 0; kc < 12; kc += 4) {
    const int kk = kc + 2 * half;
    v2f a, bv;
    a[0] = X[m * 16 + kk];
    a[1] = X[m * 16 + kk + 1];
    bv[0] = Y[kk * 16 + m];
    bv[1] = Y[(kk + 1) * 16 + m];
    acc = __builtin_amdgcn_wmma_f32_16x16x4_f32(false, a, false, bv, (short)0,
                                                acc, false, false);
  }
#pragma unroll
  for (int r = 0; r < 8; ++r) dst[(r + 8 * half) * 16 + m] = acc[r];
}

// ===========================================================================
// Kernel 1: the sequential scan. One block per batch element (B=4 blocks),
// 128 threads = 4 waves. Transition precompute (K0 = inv(L L^T) via triangular
// solves, h0, g0, plus zero-padded 16x16 Kyx/Kxy tiles) is done redundantly
// per block. Per step: 8 Gaussian integrals (GJ inverses + WMMA matmuls),
// 16 multiplies (Cholesky zeta + GJ inverse for moments), top-4-of-16.
// ===========================================================================
__global__ __launch_bounds__(128) void mgblm_scan_kernel(
    const int* __restrict__ sentences, const float* __restrict__ input_mu_w,
    const float* __restrict__ input_cho_w,
    const float* __restrict__ transition_mu,
    const float* __restrict__ transition_cho, float* __restrict__ ws_scores,
    float* __restrict__ ws_mus, float* __restrict__ ws_covs) {
  const int b = blockIdx.x;
  const int tid = threadIdx.x;
  const int lane = tid & 31;
  const int wv = tid >> 5;

  __shared__ float sK0[TCOMP][400];    // inv(trans_var), 20x20
  __shared__ float sKyxP[TCOMP][256];  // padded 16x16 Kyx tile (constant)
  __shared__ float sKxyP[TCOMP][256];  // padded 16x16 Kxy tile (constant)
  __shared__ float sh0[TCOMP][20];
  __shared__ float sg0[TCOMP];
  __shared__ float sPscore[MAXC];
  __shared__ float sPmu[MAXC][10];
  __shared__ float sPvar[MAXC][100];
  __shared__ float sBuf1[8][100];  // K1 = inv(prev var)
  __shared__ float sKxx[8][256];   // padded Kxx -> Kxx_inv (in place)
  __shared__ float sW[8][256];     // padded Kyx * Kxx_inv   (WMMA)
  __shared__ float sWK[8][256];    // padded W * Kxy         (WMMA)
  __shared__ float sKnew[8][100];  // Kyy - W*Kxy
  __shared__ float sVnew[8][100];  // inv(Knew) = p_var
  __shared__ float sHx[8][10];
  __shared__ float sHnew[8][10];
  __shared__ float sPmuN[8][10];
  __shared__ float sPs[8];
  __shared__ float sM2[16][100];  // candidate var (inv(Knew + Dinv))
  __shared__ float sMu2[16][10];
  __shared__ float sTot[16];
  __shared__ int sSel[MAXC];

  // ---- transition precompute (threads 0..TCOMP-1) ----
  // sM2 (not yet live) is reused as scratch for the 20x20 Cholesky factor.
  if (tid < TCOMP) {
    const int tc = tid;
    const float* cho = transition_cho + tc * 400;  // (20,20)
    const float* mu0 = transition_mu + tc * 20;
    float* Lm = (&sM2[0][0]) + tc * 400;  // 400-float scratch
    for (int i = 0; i < 20; ++i)
      for (int j = 0; j < 20; ++j)
        Lm[i * 20 + j] =
            (j > i) ? 0.f
                    : ((j == i) ? cho[i * 20 + j] + 1.f : cho[i * 20 + j]);
    float ldv = 0.f;
    for (int i = 0; i < 20; ++i) ldv += __logf(Lm[i * 20 + i]);
    ldv *= 2.f;  // logdet(trans_var)
    // K0 = inv(L L^T): per column, forward then back substitution
    for (int c2 = 0; c2 < 20; ++c2) {
      float yv[20], xv[20];
      for (int i = 0; i < 20; ++i) {
        float s = (i == c2) ? 1.f : 0.f;
        for (int q = 0; q < i; ++q) s -= Lm[i * 20 + q] * yv[q];
        yv[i] = s / Lm[i * 20 + i];
      }
      for (int i = 19; i >= 0; --i) {
        float s = yv[i];
        for (int q = i + 1; q < 20; ++q) s -= Lm[q * 20 + i] * xv[q];
        xv[i] = s / Lm[i * 20 + i];
      }
      for (int i = 0; i < 20; ++i) sK0[tc][i * 20 + c2] = xv[i];
    }
    float dot0 = 0.f;
    for (int i = 0; i < 20; ++i) {
      float s = 0.f;
      for (int j = 0; j < 20; ++j) s += sK0[tc][i * 20 + j] * mu0[j];
      sh0[tc][i] = s;
      dot0 += mu0[i] * s;
    }
    sg0[tc] = -0.5f * (dot0 + 20.f * LOG2PI_F + ldv);
    // padded constant tiles for the per-step WMMA matmuls
    for (int i = 0; i < 16; ++i)
      for (int j = 0; j < 16; ++j) {
        const bool in = (i < 10) && (j < 10);
        sKyxP[tc][i * 16 + j] = in ? sK0[tc][(10 + i) * 20 + j] : 0.f;
        sKxyP[tc][i * 16 + j] = in ? sK0[tc][i * 20 + 10 + j] : 0.f;
      }
  }
  // ---- init carry ----
  if (tid < MAXC) {
    sPscore[tid] = (tid == 0) ? 0.f : NEGV;
    for (int d2 = 0; d2 < 10; ++d2) sPmu[tid][d2] = 0.f;
    for (int e2 = 0; e2 < 100; ++e2)
      sPvar[tid][e2] = ((e2 / 10) == (e2 % 10)) ? 1.f : 0.f;
  }
  __syncthreads();

  for (int t = 0; t < LL; ++t) {
    const int tok = sentences[b * LL + t];
    float g1 = 0.f, ldKxx = 0.f;

    // ---- Step A: per-combo inverses (8 combos: c = k/TCOMP, tc = k%TCOMP) --
    if (tid < 8) {
      const int k = tid, c = k / TCOMP, tc = k % TCOMP;
      for (int e2 = 0; e2 < 100; ++e2) sBuf1[k][e2] = sPvar[c][e2];
      const float ld1 = gj_invert10(sBuf1[k]);  // K1, logdet(prev var)
      float d1 = 0.f;
      float h1[10];
      for (int i = 0; i < 10; ++i) {
        float s = 0.f;
        for (int j = 0; j < 10; ++j) s += sBuf1[k][i * 10 + j] * sPmu[c][j];
        h1[i] = s;
        d1 += sPmu[c][i] * s;
      }
      g1 = -0.5f * (d1 + 10.f * LOG2PI_F + ld1);
      // padded Kxx tile (zeros outside the 10x10 block)
      for (int i = 0; i < 16; ++i)
        for (int j = 0; j < 16; ++j)
          sKxx[k][i * 16 + j] =
              (i < 10 && j < 10)
                  ? sK0[tc][i * 20 + j] + sBuf1[k][i * 10 + j]
                  : 0.f;
      for (int i = 0; i < 10; ++i) sHx[k][i] = sh0[tc][i] + h1[i];
      ldKxx = gj_invert10s(sKxx[k], 16);  // sKxx now holds padded Kxx_inv
    }
    __syncthreads();

    // ---- Step B: WMMA matmuls, all 4 waves, 2 combos each ----
#pragma unroll
    for (int rep = 0; rep < 2; ++rep) {
      const int k = wv * 2 + rep;
      const int tc = k % TCOMP;
      wmma_mul16(sW[k], sKyxP[tc], sKxx[k], lane);  // W = Kyx * Kxx_inv
    }
#pragma unroll
    for (int rep = 0; rep < 2; ++rep) {
      const int k = wv * 2 + rep;
      const int tc = k % TCOMP;
      wmma_mul16(sWK[k], sW[k], sKxyP[tc], lane);  // WK = W * Kxy
    }
    __syncthreads();

    // ---- Step C: finish integral ----
    if (tid < 8) {
      const int k = tid, c = k / TCOMP, tc = k % TCOMP;
      for (int i = 0; i < 10; ++i)
        for (int j = 0; j < 10; ++j)
          sKnew[k][i * 10 + j] =
              sK0[tc][(10 + i) * 20 + (10 + j)] - sWK[k][i * 16 + j];
      for (int i = 0; i < 10; ++i) {
        float s = 0.f;
        for (int j = 0; j < 10; ++j) s += sW[k][i * 16 + j] * sHx[k][j];
        sHnew[k][i] = sh0[tc][10 + i] - s;
      }
      for (int e2 = 0; e2 < 100; ++e2) sVnew[k][e2] = sKnew[k][e2];
      const float ldKn = gj_invert10(sVnew[k]);  // p_var; logdet(K_new)
      float dmh = 0.f;
      for (int i = 0; i < 10; ++i) {
        float s = 0.f;
        for (int j = 0; j < 10; ++j) s += sVnew[k][i * 10 + j] * sHnew[k][j];
        sPmuN[k][i] = s;  // p_mu
        dmh += s * sHnew[k][i];
      }
      float qf = 0.f;
      for (int i = 0; i < 10; ++i) {
        float s = 0.f;
        for (int j = 0; j < 10; ++j) s += sKxx[k][i * 16 + j] * sHx[k][j];
        qf += sHx[k][i] * s;
      }
      const float g_marg = 0.5f * (10.f * LOG2PI_F - ldKxx + qf);
      const float log_norm = -0.5f * (dmh + 10.f * LOG2PI_F - ldKn);
      sPs[k] = sg0[tc] + g1 + g_marg - log_norm + sPscore[c];
    }
    __syncthreads();

    // ---- Step D: gaussian_multi against input comps (16 evals) ----
    if (tid < 16) {
      const int e = tid, k = e / ICOMP, ic = e % ICOMP;
      const float* imu = input_mu_w + (size_t)tok * (ICOMP * DIM) + ic * DIM;
      const float* icho = input_cho_w + (size_t)tok * (ICOMP * DIM) + ic * DIM;
      float dvar[10], dinv[10], diff[10];
      for (int d2 = 0; d2 < 10; ++d2) {
        const float cv = icho[d2];
        dvar[d2] = cv * cv;
        dinv[d2] = 1.f / (cv * cv);
        diff[d2] = sPmuN[k][d2] - imu[d2];
      }
      const float z = gauss_zeta10(sVnew[k], dvar, diff);
      sTot[e] = sPs[k] + z;
      // moments: var2 = inv(K_new + diag(dinv)); mu2 = var2*(h_new + dinv*imu)
      for (int e2 = 0; e2 < 100; ++e2) sM2[e][e2] = sKnew[k][e2];
      for (int d2 = 0; d2 < 10; ++d2) sM2[e][d2 * 10 + d2] += dinv[d2];
      gj_invert10(sM2[e]);
      float rhs[10];
      for (int d2 = 0; d2 < 10; ++d2)
        rhs[d2] = sHnew[k][d2] + dinv[d2] * imu[d2];
      for (int i = 0; i < 10; ++i) {
        float s = 0.f;
        for (int j = 0; j < 10; ++j) s += sM2[e][i * 10 + j] * rhs[j];
        sMu2[e][i] = s;
      }
    }
    __syncthreads();

    // ---- Step E: top-4 of 16 (descending, first index on ties) ----
    if (tid == 0) {
      unsigned used = 0u;
      for (int s2 = 0; s2 < MAXC; ++s2) {
        int best = 0;
        float bv = -3.4e38f;
        for (int j = 0; j < 16; ++j) {
          if (used & (1u << j)) continue;
          if (sTot[j] > bv) {
            bv = sTot[j];
            best = j;
          }
        }
        used |= 1u << best;
        sSel[s2] = best;
      }
    }
    __syncthreads();
    if (tid < MAXC) {
      const int src = sSel[tid];
      const int base = (b * LL + t) * MAXC + tid;
      const float v = sTot[src];
      sPscore[tid] = v;
      ws_scores[base] = v;
      for (int d2 = 0; d2 < 10; ++d2) {
        const float mv = sMu2[src][d2];
        sPmu[tid][d2] = mv;
        ws_mus[base * 10 + d2] = mv;
      }
      for (int e2 = 0; e2 < 100; ++e2) {
        const float cv = sM2[src][e2];
        sPvar[tid][e2] = cv;
        ws_covs[base * 100 + e2] = cv;
      }
    }
    __syncthreads();
  }
}

// ===========================================================================
// Kernel 2: output scoring. Block = (b,t) x token-chunk; 256 threads.
// Per token: 8 Cholesky-based zetas + streaming logsumexp.
// ===========================================================================
__global__ __launch_bounds__(256) void mgblm_out_kernel(
    const float* __restrict__ output_mu, const float* __restrict__ output_cho,
    const float* __restrict__ ws_scores, const float* __restrict__ ws_mus,
    const float* __restrict__ ws_covs, float* __restrict__ out) {
  const int bt = blockIdx.x;  // b*LL + t
  __shared__ float sSc[MAXC];
  __shared__ float sMu[MAXC][10];
  __shared__ float sCov[MAXC][100];
  for (int idx = threadIdx.x; idx < MAXC * 100; idx += 256)
    sCov[idx / 100][idx % 100] = ws_covs[bt * MAXC * 100 + idx];
  for (int idx = threadIdx.x; idx < MAXC * 10; idx += 256)
    sMu[idx / 10][idx % 10] = ws_mus[bt * MAXC * 10 + idx];
  if (threadIdx.x < MAXC) sSc[threadIdx.x] = ws_scores[bt * MAXC + threadIdx.x];
  __syncthreads();
  const int n = blockIdx.y * 256 + threadIdx.x;
  if (n >= NTOKN) return;
  float m = -3.4e38f, s = 0.f;
  for (int c = 0; c < MAXC; ++c) {
    for (int j = 0; j < OCOMP; ++j) {
      const float* omu = output_mu + (size_t)n * (OCOMP * DIM) + j * DIM;
      const float* ocho = output_cho + ((size_t)n * OCOMP + j) * DIM;
      float dvar[10], diff[10];
      for (int d2 = 0; d2 < 10; ++d2) {
        const float cv = ocho[d2];
        dvar[d2] = cv * cv;
        diff[d2] = sMu[c][d2] - omu[d2];
      }
      const float val = sSc[c] + gauss_zeta10(sCov[c], dvar, diff);
      if (val > m) {
        s = s * __expf(m - val) + 1.f;
        m = val;
      } else {
        s += __expf(val - m);
      }
    }
  }
  out[(size_t)bt * NTOKN + n] = m + __logf(s);
}

// ===========================================================================
// Host-side launcher. Inputs (setup_inputs order):
//  0 sentences(int32 B*L), 1 masks(unused), 2 input_mu_w, 3 input_cho_w,
//  4 transition_mu, 5 transition_cho, 6 output_mu, 7 output_cho.
// Workspace layout (floats): scores[B*L*MAXC]=256 | mus[..*10]=2560 |
// covs[..*100]=25600  (~114 KB total).
// ===========================================================================
extern "C" void kernel_launch(void* const* d_in, const int* in_sizes, int n_in,
                              void* d_out, int out_size, void* d_ws,
                              size_t ws_size, hipStream_t stream) {
  (void)in_sizes;
  (void)n_in;
  (void)out_size;
  (void)ws_size;
  const int* sentences = (const int*)d_in[0];
  const float* input_mu_w = (const float*)d_in[2];
  const float* input_cho_w = (const float*)d_in[3];
  const float* transition_mu = (const float*)d_in[4];
  const float* transition_cho = (const float*)d_in[5];
  const float* output_mu = (const float*)d_in[6];
  const float* output_cho = (const float*)d_in[7];

  float* ws = (float*)d_ws;
  float* ws_scores = ws;             // 256
  float* ws_mus = ws + 256;          // 2560
  float* ws_covs = ws + 256 + 2560;  // 25600

  mgblm_scan_kernel<<<BB, 128, 0, stream>>>(sentences, input_mu_w, input_cho_w,
                                            transition_mu, transition_cho,
                                            ws_scores, ws_mus, ws_covs);

  dim3 grid(BB * LL, (NTOKN + 255) / 256);
  mgblm_out_kernel<<<grid, 256, 0, stream>>>(output_mu, output_cho, ws_scores,
                                             ws_mus, ws_covs, (float*)d_out);
}